// Yolov1_loss_54056458387660
// MI455X (gfx1250) — compile-verified
//
#include <hip/hip_runtime.h>
#include <hip/hip_bf16.h>

#define IMG_PER_BLOCK 4
#define CH 30
#define CELLS 49
#define IMG_FLOATS (CH * CELLS)                    // 1470 floats / image / tensor
#define BLK_FLOATS (IMG_PER_BLOCK * IMG_FLOATS)    // 5880 floats
#define BLK_BYTES (BLK_FLOATS * 4)                 // 23520 bytes = 1470 x 16B
#define THREADS 256

typedef float v2f __attribute__((ext_vector_type(2)));
typedef float v8f __attribute__((ext_vector_type(8)));

// ---- CDNA5 async copy global -> LDS (tracked by ASYNCcnt) ----
__device__ __forceinline__ void async_load_b128(unsigned lds_off, const void* gaddr) {
    asm volatile("global_load_async_to_lds_b128 %0, %1, off"
                 :: "v"(lds_off), "v"((unsigned long long)gaddr)
                 : "memory");
}
__device__ __forceinline__ void wait_async0() {
    asm volatile("s_wait_asynccnt 0" ::: "memory");
}

__device__ __forceinline__ float iou_f(float cx1, float cy1, float w1, float h1,
                                       float cx2, float cy2, float w2, float h2) {
    float ax1 = cx1 - w1 * 0.5f, ay1 = cy1 - h1 * 0.5f;
    float ax2 = cx1 + w1 * 0.5f, ay2 = cy1 + h1 * 0.5f;
    float bx1 = cx2 - w2 * 0.5f, by1 = cy2 - h2 * 0.5f;
    float bx2 = cx2 + w2 * 0.5f, by2 = cy2 + h2 * 0.5f;
    float iw = fminf(ax2, bx2) - fmaxf(ax1, bx1); iw = iw > 0.0f ? iw : 0.0f;
    float ih = fminf(ay2, by2) - fmaxf(ay1, by1); ih = ih > 0.0f ? ih : 0.0f;
    float inter  = iw * ih;
    float area_a = (ax2 - ax1) * (ay2 - ay1);
    float area_b = (bx2 - bx1) * (by2 - by1);
    return inter / (area_a + area_b - inter + 1e-10f);
}

__device__ __forceinline__ float box_term(float x, float y, float w, float h,
                                          float lx, float ly, float lw, float lh) {
    float dx = x - lx, dy = y - ly;
    float dw = __builtin_sqrtf(w) - __builtin_sqrtf(lw);
    float dh = __builtin_sqrtf(h) - __builtin_sqrtf(lh);
    return dx * dx + dy * dy + dw * dw + dh * dh;
}

__global__ __launch_bounds__(THREADS)
void yolo_partial_kernel(const float* __restrict__ pred,
                         const float* __restrict__ labels,
                         float* __restrict__ ws, int n_img) {
    __shared__ float smem[2 * BLK_FLOATS];      // [pred 5880 | labels 5880] floats
    const int tid  = threadIdx.x;
    const int img0 = blockIdx.x * IMG_PER_BLOCK;

    const float* pb = pred   + (size_t)img0 * IMG_FLOATS;
    const float* lb = labels + (size_t)img0 * IMG_FLOATS;

    int vimgs = n_img - img0;
    if (vimgs > IMG_PER_BLOCK) vimgs = IMG_PER_BLOCK;
    const int vfloats = vimgs * IMG_FLOATS;
    const int chunks  = (vfloats * 4) / 16;     // full 16-byte chunks (exact for full blocks)

    // low 32 bits of a generic shared pointer == LDS byte offset
    const unsigned lds0 = (unsigned)(unsigned long long)(uintptr_t)&smem[0];

    // stream both tensors contiguously into LDS via async DMA
    for (int k = tid; k < chunks; k += THREADS) {
        async_load_b128(lds0 + (unsigned)k * 16u,             pb + k * 4);
        async_load_b128(lds0 + BLK_BYTES + (unsigned)k * 16u, lb + k * 4);
    }
    // ragged-tail fallback (never taken for N % 4 == 0)
    for (int f = chunks * 4 + tid; f < vfloats; f += THREADS) {
        smem[f]              = pb[f];
        smem[BLK_FLOATS + f] = lb[f];
    }
    wait_async0();          // this wave's async copies landed in LDS
    __syncthreads();        // all waves' copies visible

    float acc = 0.0f;
    if (tid < vimgs * CELLS) {
        const int i = tid / CELLS, j = tid % CELLS;
        const float* P = &smem[i * IMG_FLOATS + j];
        const float* L = &smem[BLK_FLOATS + i * IMG_FLOATS + j];

        const float p0 = P[0 * CELLS], p1 = P[1 * CELLS], p2 = P[2 * CELLS];
        const float p3 = P[3 * CELLS], p4 = P[4 * CELLS], p5 = P[5 * CELLS];
        const float p6 = P[6 * CELLS], p7 = P[7 * CELLS], p8 = P[8 * CELLS];
        const float p9 = P[9 * CELLS];
        const float l0 = L[0 * CELLS], l1 = L[1 * CELLS], l2 = L[2 * CELLS];
        const float l3 = L[3 * CELLS], l4 = L[4 * CELLS], l9 = L[9 * CELLS];

        const float obj   = (l4 == 1.0f) ? 1.0f : 0.0f;
        const float iou1  = iou_f(p0, p1, p2, p3, l0, l1, l2, l3);
        const float iou2  = iou_f(p5, p6, p7, p8, l0, l1, l2, l3);
        const float sel1  = (iou1 >= iou2) ? 1.0f : 0.0f;
        const float sel2  = 1.0f - sel1;

        const float d4 = p4 - l4, d9 = p9 - l9;
        const float conf = sel1 * d4 * d4 + sel2 * d9 * d9;

        const float n1 = p9 - iou1, n2 = p4 - iou2;
        const float noobj_in  = sel1 * n1 * n1 + sel2 * n2 * n2;
        const float noobj_out = p4 * p4 + p9 * p9;

        const float bbox = sel1 * box_term(p0, p1, p2, p3, l0, l1, l2, l3)
                         + sel2 * box_term(p5, p6, p7, p8, l0, l1, l2, l3);

        float cls = 0.0f;
        #pragma unroll
        for (int c = 10; c < 30; ++c) {
            float d = P[c * CELLS] - L[c * CELLS];
            cls += d * d;
        }

        acc = 5.0f * obj * bbox
            + 0.1f * (obj * noobj_in + (1.0f - obj) * noobj_out)
            + obj * conf
            + obj * cls;
    }

    // ---- wave32 reduction via V_WMMA_F32_16X16X4_F32 (ones trick) ----
    // A[m][0] = acc(lane m), A[m][2] = acc(lane m+16); B = ones(4x16)
    // => D[m][n] = acc_m + acc_{m+16}; sum 8 D regs + shfl_xor(16) = wave sum.
    v2f a; a[0] = acc;  a[1] = 0.0f;
    v2f b; b[0] = 1.0f; b[1] = 1.0f;
    v8f c = {0.0f, 0.0f, 0.0f, 0.0f, 0.0f, 0.0f, 0.0f, 0.0f};
    c = __builtin_amdgcn_wmma_f32_16x16x4_f32(false, a, false, b, (short)0, c, false, false);
    float s = c[0] + c[1] + c[2] + c[3] + c[4] + c[5] + c[6] + c[7];
    float wavetot = s + __shfl_xor(s, 16, 32);

    __shared__ float wred[THREADS / 32];
    if ((tid & 31) == 0) wred[tid >> 5] = wavetot;
    __syncthreads();
    if (tid == 0) {
        float bsum = 0.0f;
        #pragma unroll
        for (int w = 0; w < THREADS / 32; ++w) bsum += wred[w];
        ws[blockIdx.x] = bsum;
    }
}

__global__ __launch_bounds__(256)
void yolo_final_kernel(const float* __restrict__ ws, int nblocks,
                       float* __restrict__ out, double invN) {
    const int tid = threadIdx.x;
    double acc = 0.0;
    for (int i = tid; i < nblocks; i += 256) acc += (double)ws[i];
    for (int off = 16; off > 0; off >>= 1)
        acc += __shfl_down(acc, off, 32);
    __shared__ double wr[8];
    if ((tid & 31) == 0) wr[tid >> 5] = acc;
    __syncthreads();
    if (tid == 0) {
        double t = 0.0;
        #pragma unroll
        for (int w = 0; w < 8; ++w) t += wr[w];
        out[0] = (float)(t * invN);
    }
}

extern "C" void kernel_launch(void* const* d_in, const int* in_sizes, int n_in,
                              void* d_out, int out_size, void* d_ws, size_t ws_size,
                              hipStream_t stream) {
    const float* pred   = (const float*)d_in[0];
    const float* labels = (const float*)d_in[1];
    float* ws  = (float*)d_ws;
    float* out = (float*)d_out;

    const int n_img   = in_sizes[0] / IMG_FLOATS;                     // 65536
    const int nblocks = (n_img + IMG_PER_BLOCK - 1) / IMG_PER_BLOCK;  // 16384

    yolo_partial_kernel<<<nblocks, THREADS, 0, stream>>>(pred, labels, ws, n_img);
    yolo_final_kernel<<<1, 256, 0, stream>>>(ws, nblocks, out, 1.0 / (double)n_img);
}